// MinibatchDiscrimination_55035710931183
// MI455X (gfx1250) — compile-verified
//
#include <hip/hip_runtime.h>

typedef __attribute__((ext_vector_type(2))) float v2f;
typedef __attribute__((ext_vector_type(8))) float v8f;

#define BATCH 128
#define IN_F  512
#define OUT_F 128
#define KDIM  32
#define NDIM  (OUT_F * KDIM)   /* 4096 */
#define OUTW  (IN_F + OUT_F)   /* 640  */

// ---------------------------------------------------------------------------
// Kernel 1: m[B, N] = x[B, K] @ W[K, N]   (B=128, K=512, N=4096), all f32.
// One wave computes a 16x64 tile with V_WMMA_F32_16X16X4_F32.
// Block = 128 threads = 4 waves -> 16 x 256 tile. Grid = (4096/256, 128/16).
// Fragment layouts (CDNA5 ISA 7.12.2):
//   A 16x4 f32 : lane&15 = M, lane>>4 selects K pair {0,1}/{2,3}; VGPR0,1 = the pair
//   B 4x16 f32 : lane&15 = N, VGPRv + 2*(lane>>4) = K
//   C/D 16x16  : lane&15 = N, M = v + 8*(lane>>4)
// ---------------------------------------------------------------------------
__global__ __launch_bounds__(128) void gemm_wmma_f32(const float* __restrict__ x,
                                                     const float* __restrict__ W,
                                                     float* __restrict__ m)
{
    const int lane = threadIdx.x & 31;
    const int wave = threadIdx.x >> 5;
    const int half = lane >> 4;
    const int row  = lane & 15;

    const int m0 = blockIdx.y * 16;
    const int n0 = blockIdx.x * 256 + wave * 64;

    v8f c0 = {}, c1 = {}, c2 = {}, c3 = {};

    const float* __restrict__ xrow = x + (size_t)(m0 + row) * IN_F;

    #pragma unroll 2
    for (int k4 = 0; k4 < IN_F; k4 += 4) {
        const int k = k4 + 2 * half;

        v2f a;
        a.x = xrow[k];
        a.y = xrow[k + 1];

        const float* __restrict__ w0 = W + (size_t)k * NDIM + n0 + row;
        const float* __restrict__ w1 = w0 + NDIM;

        v2f b0, b1, b2, b3;
        b0.x = w0[0];   b0.y = w1[0];
        b1.x = w0[16];  b1.y = w1[16];
        b2.x = w0[32];  b2.y = w1[32];
        b3.x = w0[48];  b3.y = w1[48];

        c0 = __builtin_amdgcn_wmma_f32_16x16x4_f32(false, a, false, b0, (short)0, c0, false, false);
        c1 = __builtin_amdgcn_wmma_f32_16x16x4_f32(false, a, false, b1, (short)0, c1, false, false);
        c2 = __builtin_amdgcn_wmma_f32_16x16x4_f32(false, a, false, b2, (short)0, c2, false, false);
        c3 = __builtin_amdgcn_wmma_f32_16x16x4_f32(false, a, false, b3, (short)0, c3, false, false);
    }

    float* __restrict__ mout = m + (size_t)m0 * NDIM + n0;
    #pragma unroll
    for (int v = 0; v < 8; ++v) {
        const int M = v + 8 * half;
        mout[(size_t)M * NDIM +      row] = c0[v];
        mout[(size_t)M * NDIM + 16 + row] = c1[v];
        mout[(size_t)M * NDIM + 32 + row] = c2[v];
        mout[(size_t)M * NDIM + 48 + row] = c3[v];
    }
}

// ---------------------------------------------------------------------------
// Kernel 2: one block per output feature o.
//   Stage m[:, o, 0:32] (128 rows x 32 floats = 16 KB) into LDS.
//   Thread j: norm(i,j) = sum_k |m[i,o,k]-m[j,o,k]|; out[j,512+o] = sum_i exp(-norm) - 1
// ---------------------------------------------------------------------------
__global__ __launch_bounds__(128) void pairwise_kernel(const float* __restrict__ m,
                                                       float* __restrict__ out)
{
    __shared__ __align__(16) float sm[BATCH * KDIM]; // 16 KB

    const int o   = blockIdx.x;
    const int tid = threadIdx.x;

    float4* __restrict__ smv = (float4*)sm;
    #pragma unroll
    for (int idx = tid; idx < BATCH * (KDIM / 4); idx += 128) {
        const int r = idx >> 3;  // row i
        const int q = idx & 7;   // float4 within row
        smv[idx] = *(const float4*)(m + (size_t)r * NDIM + o * KDIM + q * 4);
    }
    __syncthreads();

    const int j = tid;
    float4 mj[8];
    #pragma unroll
    for (int q = 0; q < 8; ++q) mj[q] = smv[j * 8 + q];

    float sum = 0.0f;
    for (int i = 0; i < BATCH; ++i) {
        float norm = 0.0f;
        #pragma unroll
        for (int q = 0; q < 8; ++q) {
            const float4 a = smv[i * 8 + q];
            norm += fabsf(a.x - mj[q].x) + fabsf(a.y - mj[q].y)
                  + fabsf(a.z - mj[q].z) + fabsf(a.w - mj[q].w);
        }
        sum += __expf(-norm);
    }
    out[(size_t)j * OUTW + IN_F + o] = sum - 1.0f; // subtract self term exp(0)
}

// ---------------------------------------------------------------------------
// Kernel 3: out[:, 0:512] = x   (B128 vectorized copy)
// ---------------------------------------------------------------------------
__global__ __launch_bounds__(128) void copy_x_kernel(const float* __restrict__ x,
                                                     float* __restrict__ out)
{
    const int r = blockIdx.x;
    const int t = threadIdx.x;
    const float4 v = *(const float4*)(x + (size_t)r * IN_F + t * 4);
    *(float4*)(out + (size_t)r * OUTW + t * 4) = v;
}

extern "C" void kernel_launch(void* const* d_in, const int* in_sizes, int n_in,
                              void* d_out, int out_size, void* d_ws, size_t ws_size,
                              hipStream_t stream) {
    const float* x = (const float*)d_in[0];          // [128, 512]
    const float* T = (const float*)d_in[1];          // [512, 4096] row-major (in, out*k)
    float*       out = (float*)d_out;                // [128, 640]
    float*       m   = (float*)d_ws;                 // [128, 4096] scratch, 2 MB

    dim3 ggrid(NDIM / 256, BATCH / 16);              // (16, 8)
    gemm_wmma_f32<<<ggrid, 128, 0, stream>>>(x, T, m);
    pairwise_kernel<<<OUT_F, 128, 0, stream>>>(m, out);
    copy_x_kernel<<<BATCH, 128, 0, stream>>>(x, out);
}